// RPN2RoI_23527830848122
// MI455X (gfx1250) — compile-verified
//
#include <hip/hip_runtime.h>
#include <stdint.h>

// ---------------------------------------------------------------------------
// RPN -> RoI post-processing for MI455X (gfx1250, wave32).
// Pipeline: radix-select top-k per (image,level) on raw logits -> bitonic sort
// -> anchor decode -> per-level tiled greedy NMS (WMMA rank-2 union term)
// -> 5-way merge per image.
// ---------------------------------------------------------------------------

typedef __attribute__((ext_vector_type(2))) float v2f;
typedef __attribute__((ext_vector_type(8))) float v8f;

#define NIMG    8
#define NLVL    5
#define NSEG    (NIMG * NLVL)
#define CAP     2048          // per-segment slot capacity (>= max k, power of 2)
#define MAX_OUT 1000
#define NMS_THR 0.7f
#define MAX_RATIO 4.135166556742356f   // |ln(16/1000)|

__device__ __constant__ int dH[NLVL]      = {208, 104, 52, 26, 13};
__device__ __constant__ int dW[NLVL]      = {336, 168, 84, 42, 21};
__device__ __constant__ int dSTRIDE[NLVL] = {4, 8, 16, 32, 64};
__device__ __constant__ int dELEMS[NLVL]  = {209664, 52416, 13104, 3276, 819}; // 3*H*W
__device__ __constant__ int dK[NLVL]      = {2000, 2000, 2000, 2000, 819};
__device__ __constant__ int dNPAD[NLVL]   = {2000, 2000, 2000, 2000, 832};     // ceil(k/16)*16
// ratios (0.5,1,2): h_r=sqrt(r), w_r=1/sqrt(r)
__device__ __constant__ float dHR[3] = {0.70710678118f, 1.0f, 1.41421356237f};
__device__ __constant__ float dWR[3] = {1.41421356237f, 1.0f, 0.70710678118f};

// workspace byte offsets
#define OFF_HIST   0u          // NSEG*256 u32   = 40960
#define OFF_PREFIX 40960u      // NSEG u32
#define OFF_CGT    41120u      // NSEG u32 (count strictly greater than threshold)
#define OFF_KREM   41280u      // NSEG u32 (remaining k within current bin / ties needed)
#define OFF_CNTGT  41440u      // NSEG u32 (compaction cursor, > threshold)
#define OFF_CNTEQ  41600u      // NSEG u32 (compaction cursor, == threshold)
#define OFF_SEL    65536u      // NSEG*CAP u64   = 655360
#define OFF_BOX    720896u     // NSEG*CAP*4 f32 = 1310720
#define OFF_AREA   2031616u    // NSEG*CAP f32   = 327680
#define OFF_SCORE  2359296u    // NSEG*CAP f32   = 327680
#define OFF_KEEP   2686976u    // NSEG*CAP u32   = 327680  (end ~3.0MB)

__device__ __forceinline__ unsigned fkey(float f) {
    unsigned u = __float_as_uint(f);
    return u ^ ((u >> 31) ? 0xFFFFFFFFu : 0x80000000u);  // order-preserving (desc on larger key)
}
__device__ __forceinline__ float unfkey(unsigned k) {
    unsigned u = (k & 0x80000000u) ? (k ^ 0x80000000u) : ~k;
    return __uint_as_float(u);
}

// ---------------------------------------------------------------------------
__global__ void initKernel(unsigned* hist, unsigned* prefix, unsigned* cgt,
                           unsigned* krem, unsigned* cntgt, unsigned* cnteq,
                           unsigned long long* sel, float* out) {
    int g = blockIdx.x * blockDim.x + threadIdx.x;
    int n = gridDim.x * blockDim.x;
    for (int i = g; i < NSEG * 256; i += n) hist[i] = 0u;
    for (int i = g; i < NSEG; i += n) {
        prefix[i] = 0u; cgt[i] = 0u; cntgt[i] = 0u; cnteq[i] = 0u;
        krem[i] = (unsigned)dK[i % NLVL];
    }
    for (int i = g; i < NSEG * CAP; i += n) sel[i] = 0ull;
    for (int i = g; i < NIMG * MAX_OUT * 5; i += n) out[i] = 0.0f;  // boxes + scores
}

// ---------------------------------------------------------------------------
// Radix-select histogram pass p (8 bits per pass, MSB first), grid (819, NSEG)
__global__ void histKernel(const float* __restrict__ c0, const float* __restrict__ c1,
                           const float* __restrict__ c2, const float* __restrict__ c3,
                           const float* __restrict__ c4,
                           unsigned* __restrict__ hist, const unsigned* __restrict__ prefix,
                           int p) {
    __shared__ unsigned lh[256];
    int seg = blockIdx.y, img = seg / NLVL, lvl = seg % NLVL;
    lh[threadIdx.x] = 0u;
    __syncthreads();
    int elems = dELEMS[lvl];
    int t = blockIdx.x * 256 + threadIdx.x;
    const float* cp = (lvl == 0) ? c0 : (lvl == 1) ? c1 : (lvl == 2) ? c2 : (lvl == 3) ? c3 : c4;
    if (t < elems) {
        __builtin_prefetch(&cp[img * elems + t + 8192], 0, 1);   // global_prefetch_b8
        float v = cp[img * elems + t];
        unsigned key = fkey(v);
        unsigned pre = prefix[seg];
        bool ok = (p == 0) || (((key ^ pre) >> (32 - 8 * p)) == 0);
        if (ok) atomicAdd(&lh[(key >> (24 - 8 * p)) & 255u], 1u);
    }
    __syncthreads();
    unsigned v = lh[threadIdx.x];
    if (v) atomicAdd(&hist[seg * 256 + threadIdx.x], v);
}

// One thread per segment: pick bin containing the k-th largest, refine prefix.
__global__ void scanKernel(unsigned* hist, unsigned* prefix, unsigned* cgt,
                           unsigned* krem, int p) {
    int s = blockIdx.x * blockDim.x + threadIdx.x;
    if (s >= NSEG) return;
    unsigned need = krem[s];
    unsigned cum = 0u, above = 0u;
    int bin = 0; bool found = false;
    for (int b = 255; b >= 0; --b) {
        unsigned h = hist[s * 256 + b];
        hist[s * 256 + b] = 0u;                 // re-zero for next pass
        if (!found) {
            if (cum + h >= need) { found = true; bin = b; above = cum; }
            cum += h;
        }
    }
    prefix[s] |= (unsigned)bin << (24 - 8 * p);
    cgt[s] += above;
    krem[s] = need - above;
}

// Compaction: keys > T fill [0, cgt); keys == T fill [cgt, k). grid (819, NSEG)
__global__ void compactKernel(const float* __restrict__ c0, const float* __restrict__ c1,
                              const float* __restrict__ c2, const float* __restrict__ c3,
                              const float* __restrict__ c4,
                              const unsigned* __restrict__ prefix, const unsigned* __restrict__ cgt,
                              const unsigned* __restrict__ krem,
                              unsigned* cntgt, unsigned* cnteq,
                              unsigned long long* __restrict__ sel) {
    int seg = blockIdx.y, img = seg / NLVL, lvl = seg % NLVL;
    int elems = dELEMS[lvl];
    int t = blockIdx.x * 256 + threadIdx.x;
    if (t >= elems) return;
    const float* cp = (lvl == 0) ? c0 : (lvl == 1) ? c1 : (lvl == 2) ? c2 : (lvl == 3) ? c3 : c4;
    unsigned key = fkey(cp[img * elems + t]);
    unsigned T = prefix[seg];
    int pos = -1;
    if (key > T) {
        pos = (int)atomicAdd(&cntgt[seg], 1u);
    } else if (key == T) {
        unsigned q = atomicAdd(&cnteq[seg], 1u);
        if (q < krem[seg]) pos = (int)(cgt[seg] + q);
    }
    if (pos >= 0) {
        int HW = dH[lvl] * dW[lvl];
        int a = t / HW, yx = t % HW;
        unsigned si = (unsigned)(yx * 3 + a);   // index in reference score layout
        // tie-break: smaller si sorts first under descending u64 -> store ~si low
        sel[seg * CAP + pos] = ((unsigned long long)key << 32) | (unsigned long long)(~si);
    }
}

// ---------------------------------------------------------------------------
// Per-segment bitonic sort, descending, 2048 u64 in LDS. 40 blocks x 512.
__global__ void sortKernel(unsigned long long* __restrict__ sel) {
    __shared__ unsigned long long sh[CAP];
    int seg = blockIdx.x;
    for (int i = threadIdx.x; i < CAP; i += blockDim.x) sh[i] = sel[seg * CAP + i];
    __syncthreads();
    for (int k = 2; k <= CAP; k <<= 1) {
        for (int j = k >> 1; j > 0; j >>= 1) {
            for (int i = threadIdx.x; i < CAP; i += blockDim.x) {
                int ixj = i ^ j;
                if (ixj > i) {
                    unsigned long long a = sh[i], b = sh[ixj];
                    bool asc = (i & k) != 0;            // descending overall
                    bool sw = asc ? (a > b) : (a < b);
                    if (sw) { sh[i] = b; sh[ixj] = a; }
                }
            }
            __syncthreads();
        }
    }
    for (int i = threadIdx.x; i < CAP; i += blockDim.x) sel[seg * CAP + i] = sh[i];
}

// ---------------------------------------------------------------------------
// Anchor + DeltaXYWH decode for each sorted slot.
__global__ void decodeKernel(const unsigned long long* __restrict__ sel,
                             const float* __restrict__ r0, const float* __restrict__ r1,
                             const float* __restrict__ r2, const float* __restrict__ r3,
                             const float* __restrict__ r4,
                             const int* __restrict__ ih, const int* __restrict__ iw,
                             float* __restrict__ box, float* __restrict__ area,
                             float* __restrict__ score) {
    int g = blockIdx.x * blockDim.x + threadIdx.x;
    if (g >= NSEG * CAP) return;
    int seg = g / CAP, slot = g % CAP;
    int img = seg / NLVL, lvl = seg % NLVL;
    if (slot >= dK[lvl]) {
        box[g * 4 + 0] = 0.f; box[g * 4 + 1] = 0.f; box[g * 4 + 2] = 0.f; box[g * 4 + 3] = 0.f;
        area[g] = 0.f; score[g] = 0.f;
        return;
    }
    unsigned long long e = sel[g];
    unsigned key = (unsigned)(e >> 32);
    unsigned si  = ~(unsigned)(e & 0xFFFFFFFFull);
    float logit = unfkey(key);
    float sc = 1.0f / (1.0f + expf(-logit));

    int W = dW[lvl], H = dH[lvl];
    int a = (int)(si % 3u);
    int pos = (int)(si / 3u);
    int y = pos / W, x = pos % W;
    float st = (float)dSTRIDE[lvl];
    float pw = 8.0f * st * dWR[a];
    float ph = 8.0f * st * dHR[a];
    float px = (float)x * st;
    float py = (float)y * st;

    const float* rp = (lvl == 0) ? r0 : (lvl == 1) ? r1 : (lvl == 2) ? r2 : (lvl == 3) ? r3 : r4;
    int HW = H * W;
    long base = (long)img * 12 * HW + (long)(4 * a) * HW + (long)y * W + x;
    float dx = rp[base];
    float dy = rp[base + HW];
    float dw = fminf(fmaxf(rp[base + 2 * HW], -MAX_RATIO), MAX_RATIO);
    float dh = fminf(fmaxf(rp[base + 3 * HW], -MAX_RATIO), MAX_RATIO);

    float gx = px + pw * dx;
    float gy = py + ph * dy;
    float gw = pw * expf(dw);
    float gh = ph * expf(dh);
    float Wf = (float)iw[0], Hf = (float)ih[0];
    float x1 = fminf(fmaxf(gx - gw * 0.5f, 0.f), Wf);
    float y1 = fminf(fmaxf(gy - gh * 0.5f, 0.f), Hf);
    float x2 = fminf(fmaxf(gx + gw * 0.5f, 0.f), Wf);
    float y2 = fminf(fmaxf(gy + gh * 0.5f, 0.f), Hf);
    box[g * 4 + 0] = x1; box[g * 4 + 1] = y1; box[g * 4 + 2] = x2; box[g * 4 + 3] = y2;
    area[g] = (x2 - x1) * (y2 - y1);
    score[g] = sc;
}

// ---------------------------------------------------------------------------
// Per-segment greedy NMS, one wave32 per block, 16x16 tiles. The pairwise
// union base (area_i + area_j) is a rank-2 outer sum computed with
// v_wmma_f32_16x16x4_f32: A = [area_i, 1, 0, 0] (16x4), B = [1; area_j; 0; 0].
// IoU threshold test uses the multiply form (inter > thr*max(union,1e-6)),
// which is order-equivalent to the reference's divide form since the RHS
// denominator is strictly positive.
__global__ __launch_bounds__(32) void nmsKernel(const float* __restrict__ box,
                                                const float* __restrict__ area,
                                                unsigned* __restrict__ keep) {
    __shared__ float bX1[CAP], bY1[CAP], bX2[CAP], bY2[CAP], bA[CAP];
    __shared__ unsigned kp[CAP];
    int seg = blockIdx.x, lvl = seg % NLVL;
    int kK = dK[lvl], nPad = dNPAD[lvl];
    int lane = threadIdx.x;
    for (int i = lane; i < CAP; i += 32) {
        long g = (long)seg * CAP + i;
        bX1[i] = box[g * 4 + 0]; bY1[i] = box[g * 4 + 1];
        bX2[i] = box[g * 4 + 2]; bY2[i] = box[g * 4 + 3];
        bA[i] = area[g];
        kp[i] = (i < kK) ? 1u : 0u;
    }
    __syncthreads();

    bool lowHalf = (lane < 16);
    int lj = lane & 15;

    for (int ti = 0; ti < nPad; ti += 16) {
        // --- intra-tile serial suppression (greedy order within the tile) ---
        for (int q = 0; q < 16; ++q) {
            int i = ti + q;
            unsigned ki = kp[i];
            // unconditional LDS reads; predicate only the write
            int j = ti + lj;
            float xx1 = fmaxf(bX1[i], bX1[j]), yy1 = fmaxf(bY1[i], bY1[j]);
            float xx2 = fminf(bX2[i], bX2[j]), yy2 = fminf(bY2[i], bY2[j]);
            float inter = fmaxf(xx2 - xx1, 0.f) * fmaxf(yy2 - yy1, 0.f);
            float un = fmaxf(bA[i] + bA[j] - inter, 1e-6f);
            bool sup = inter > NMS_THR * un;     // == (inter/un > thr), un > 0
            if (ki && lowHalf && lane > q && sup) kp[j] = 0u;
            __syncthreads();
        }
        // --- hoist the 8 row boxes + final keep flags for this lane ---
        int ibase = ti + (lowHalf ? 0 : 8);
        float rx1[8], ry1[8], rx2[8], ry2[8];
        unsigned rk[8];
        #pragma unroll
        for (int kk = 0; kk < 8; ++kk) {
            int i = ibase + kk;
            rx1[kk] = bX1[i]; ry1[kk] = bY1[i];
            rx2[kk] = bX2[i]; ry2[kk] = bY2[i];
            rk[kk] = kp[i];
        }
        float aRow = bA[ti + lj];                // area_i for A operand (lanes 0-15)
        v2f Am, Bconst;
        Am.x = lowHalf ? aRow : 0.f;             // A: K=0 row = area_i (K=2 -> 0)
        Am.y = lowHalf ? 1.0f : 0.f;             // A: K=1 row = 1      (K=3 -> 0)
        Bconst.x = lowHalf ? 1.0f : 0.f;         // B[0][j] = 1

        // --- suppress all later column tiles; kp[ti..ti+15] is now final ---
        for (int tj = ti + 16; tj < nPad; tj += 16) {
            int j = tj + lj;
            float aj  = bA[j];                   // same value feeds Bm.y for lanes 0-15
            float x1j = bX1[j], y1j = bY1[j], x2j = bX2[j], y2j = bY2[j];
            v2f Bm;
            Bm.x = Bconst.x;
            Bm.y = lowHalf ? aj : 0.f;           // B[1][j] = area_j
            v8f U = {};
#if defined(__has_builtin) && __has_builtin(__builtin_amdgcn_wmma_f32_16x16x4_f32)
            U = __builtin_amdgcn_wmma_f32_16x16x4_f32(false, Am, false, Bm,
                                                      (short)0, U, false, false);
#else
            #pragma unroll
            for (int kk = 0; kk < 8; ++kk) U[kk] = bA[ibase + kk] + aj;
#endif
            unsigned sup = 0u;
            #pragma unroll
            for (int kk = 0; kk < 8; ++kk) {
                // C layout: VGPR kk, lane l -> M = kk + 8*(l>=16), N = l&15
                float xx1 = fmaxf(rx1[kk], x1j), yy1 = fmaxf(ry1[kk], y1j);
                float xx2 = fminf(rx2[kk], x2j), yy2 = fminf(ry2[kk], y2j);
                float inter = fmaxf(xx2 - xx1, 0.f) * fmaxf(yy2 - yy1, 0.f);
                float un = fmaxf(U[kk] - inter, 1e-6f);
                sup |= (rk[kk] && (inter > NMS_THR * un)) ? 1u : 0u;
            }
            unsigned o = __shfl_xor(sup, 16, 32);   // combine the two lane halves
            if (lowHalf && (sup | o)) kp[j] = 0u;
        }
        __syncthreads();
    }
    for (int i = lane; i < CAP; i += 32)
        keep[seg * CAP + i] = (i < kK) ? kp[i] : 0u;
}

// ---------------------------------------------------------------------------
// Per-image 5-way merge of kept, per-level-sorted lists; take first 1000.
__global__ void mergeKernel(const unsigned long long* __restrict__ sel,
                            const float* __restrict__ box, const float* __restrict__ score,
                            const unsigned* __restrict__ keep, float* __restrict__ out) {
    int img = blockIdx.x;
    if (threadIdx.x != 0) return;
    int cur[NLVL];
    for (int l = 0; l < NLVL; ++l) {
        int c = 0, kK = dK[l], s = img * NLVL + l;
        while (c < kK && keep[s * CAP + c] == 0u) ++c;
        cur[l] = c;
    }
    float* ob = out + (long)img * MAX_OUT * 4;
    float* os = out + (long)NIMG * MAX_OUT * 4 + (long)img * MAX_OUT;
    for (int r = 0; r < MAX_OUT; ++r) {
        int best = -1; unsigned long long bk = 0ull;
        for (int l = 0; l < NLVL; ++l) {
            if (cur[l] < dK[l]) {
                unsigned long long kk = sel[(img * NLVL + l) * CAP + cur[l]];
                if (best < 0 || kk > bk) { best = l; bk = kk; }  // ties -> lowest level
            }
        }
        if (best < 0) break;
        int s = img * NLVL + best, c = cur[best];
        long g = (long)s * CAP + c;
        ob[r * 4 + 0] = box[g * 4 + 0]; ob[r * 4 + 1] = box[g * 4 + 1];
        ob[r * 4 + 2] = box[g * 4 + 2]; ob[r * 4 + 3] = box[g * 4 + 3];
        os[r] = score[g];
        ++c;
        while (c < dK[best] && keep[(long)s * CAP + c] == 0u) ++c;
        cur[best] = c;
    }
}

// ---------------------------------------------------------------------------
extern "C" void kernel_launch(void* const* d_in, const int* in_sizes, int n_in,
                              void* d_out, int out_size, void* d_ws, size_t ws_size,
                              hipStream_t stream) {
    (void)in_sizes; (void)n_in; (void)out_size; (void)ws_size;
    // setup_inputs() dict order: class_0, reg_0, class_1, reg_1, ..., image_h, image_w
    const float* c0 = (const float*)d_in[0];
    const float* r0 = (const float*)d_in[1];
    const float* c1 = (const float*)d_in[2];
    const float* r1 = (const float*)d_in[3];
    const float* c2 = (const float*)d_in[4];
    const float* r2 = (const float*)d_in[5];
    const float* c3 = (const float*)d_in[6];
    const float* r3 = (const float*)d_in[7];
    const float* c4 = (const float*)d_in[8];
    const float* r4 = (const float*)d_in[9];
    const int* ih = (const int*)d_in[10];
    const int* iw = (const int*)d_in[11];
    float* out = (float*)d_out;

    char* ws = (char*)d_ws;
    unsigned* hist   = (unsigned*)(ws + OFF_HIST);
    unsigned* prefix = (unsigned*)(ws + OFF_PREFIX);
    unsigned* cgt    = (unsigned*)(ws + OFF_CGT);
    unsigned* krem   = (unsigned*)(ws + OFF_KREM);
    unsigned* cntgt  = (unsigned*)(ws + OFF_CNTGT);
    unsigned* cnteq  = (unsigned*)(ws + OFF_CNTEQ);
    unsigned long long* sel = (unsigned long long*)(ws + OFF_SEL);
    float* box   = (float*)(ws + OFF_BOX);
    float* area  = (float*)(ws + OFF_AREA);
    float* score = (float*)(ws + OFF_SCORE);
    unsigned* keep = (unsigned*)(ws + OFF_KEEP);

    initKernel<<<256, 256, 0, stream>>>(hist, prefix, cgt, krem, cntgt, cnteq, sel, out);

    dim3 segGrid(819, NSEG);  // 819 = ceil(209664/256), smaller levels early-exit
    for (int p = 0; p < 4; ++p) {
        histKernel<<<segGrid, 256, 0, stream>>>(c0, c1, c2, c3, c4, hist, prefix, p);
        scanKernel<<<1, 64, 0, stream>>>(hist, prefix, cgt, krem, p);
    }
    compactKernel<<<segGrid, 256, 0, stream>>>(c0, c1, c2, c3, c4, prefix, cgt, krem,
                                               cntgt, cnteq, sel);
    sortKernel<<<NSEG, 512, 0, stream>>>(sel);
    decodeKernel<<<(NSEG * CAP + 255) / 256, 256, 0, stream>>>(sel, r0, r1, r2, r3, r4,
                                                               ih, iw, box, area, score);
    nmsKernel<<<NSEG, 32, 0, stream>>>(box, area, keep);
    mergeKernel<<<NIMG, 32, 0, stream>>>(sel, box, score, keep, out);
}